// TagCountAE_25984552140927
// MI455X (gfx1250) — compile-verified
//
#include <hip/hip_runtime.h>
#include <hip/hip_bf16.h>
#include <math.h>

typedef __attribute__((ext_vector_type(2))) float v2f;
typedef __attribute__((ext_vector_type(8))) float v8f;

#define NUM_TAGS 20000
#define BATCH 64
#define NFEAT 128
#define NTAGROW 16
#define H4 1024
#define HDIM 256
#define YW (NUM_TAGS + 1)   // 20001

#define GEMM_NW 8                                   // waves per block (split-K)
#define GEMM_SMEM ((GEMM_NW - 1) * 4 * 8 * 32 * 4)  // spill area for waves 1..NW-1

__device__ __forceinline__ float gelu_exact(float x) {
    return 0.5f * x * (1.0f + erff(x * 0.7071067811865475f));
}

// ---------------- zero fill ----------------
__global__ void zero_f32_kernel(float* __restrict__ p, int n) {
    int i = blockIdx.x * blockDim.x + threadIdx.x;
    if (i < n) p[i] = 0.0f;
}

// ---------------- dedup scatter-count ----------------
// one thread per (b, f) row of 16 tags; duplicates within a row collapse to 1.
// atomicAdd of 1.0f sums are exact small integers -> order-independent result.
__global__ void scatter_count_kernel(const int* __restrict__ tags,
                                     float* __restrict__ cnt) {
    int idx = blockIdx.x * blockDim.x + threadIdx.x;   // 0 .. B*F-1
    if (idx >= BATCH * NFEAT) return;
    int b = idx >> 7;                                  // F = 128
    const int* row = tags + (size_t)idx * NTAGROW;
    int t[NTAGROW];
#pragma unroll
    for (int i = 0; i < NTAGROW; ++i) t[i] = row[i];
#pragma unroll
    for (int i = 0; i < NTAGROW; ++i) {
        bool first = true;
#pragma unroll
        for (int j = 0; j < NTAGROW; ++j)
            if (j < i && t[j] == t[i]) first = false;
        if (first) atomicAdd(cnt + (size_t)b * NUM_TAGS + t[i], 1.0f);
    }
}

// ---------------- LayerNorm(20000) + concat -> y ----------------
__global__ void ln_y_kernel(const float* __restrict__ cnt,
                            const float* __restrict__ fc,
                            const float* __restrict__ g,
                            const float* __restrict__ bb,
                            float* __restrict__ y) {
    __shared__ float red[256];
    int b = blockIdx.x, t = threadIdx.x;
    const float* row = cnt + (size_t)b * NUM_TAGS;
    float s = 0.0f;
    for (int i = t; i < NUM_TAGS; i += 256) s += row[i];
    red[t] = s; __syncthreads();
    for (int off = 128; off > 0; off >>= 1) {
        if (t < off) red[t] += red[t + off];
        __syncthreads();
    }
    float mu = red[0] * (1.0f / NUM_TAGS);
    __syncthreads();
    float v = 0.0f;
    for (int i = t; i < NUM_TAGS; i += 256) { float d = row[i] - mu; v += d * d; }
    red[t] = v; __syncthreads();
    for (int off = 128; off > 0; off >>= 1) {
        if (t < off) red[t] += red[t + off];
        __syncthreads();
    }
    float rstd = rsqrtf(red[0] * (1.0f / NUM_TAGS) + 1e-5f);
    float* yr = y + (size_t)b * YW;
    if (t == 0) yr[0] = fc[b] * 0.01f;
    for (int i = t; i < NUM_TAGS; i += 256)
        yr[1 + i] = (row[i] - mu) * rstd * g[i] + bb[i];
}

// ---------------- LayerNorm(1024) ----------------
__global__ void ln_h_kernel(const float* __restrict__ x,
                            const float* __restrict__ g,
                            const float* __restrict__ bb,
                            float* __restrict__ o) {
    __shared__ float red[256];
    int r = blockIdx.x, t = threadIdx.x;
    const float* xr = x + (size_t)r * H4;
    float s = 0.0f;
    for (int i = t; i < H4; i += 256) s += xr[i];
    red[t] = s; __syncthreads();
    for (int off = 128; off > 0; off >>= 1) {
        if (t < off) red[t] += red[t + off];
        __syncthreads();
    }
    float mu = red[0] * (1.0f / H4);
    __syncthreads();
    float v = 0.0f;
    for (int i = t; i < H4; i += 256) { float d = xr[i] - mu; v += d * d; }
    red[t] = v; __syncthreads();
    for (int off = 128; off > 0; off >>= 1) {
        if (t < off) red[t] += red[t + off];
        __syncthreads();
    }
    float rstd = rsqrtf(red[0] * (1.0f / H4) + 1e-5f);
    float* orow = o + (size_t)r * H4;
    for (int i = t; i < H4; i += 256)
        orow[i] = (xr[i] - mu) * rstd * g[i] + bb[i];
}

// ---- WMMA fp32 GEMM with in-block split-K ----------------------------------
// C[64,N] = act(A[64,K] @ B[K,N] + bias)
// Block = GEMM_NW waves; each block owns one 16-wide N panel of B (weights are
// streamed from HBM exactly once). Waves take interleaved K-chunks of 4 (so
// the block touches 4*NW consecutive weight rows at a time) and accumulate
// 4 m-tile partials in registers. Strength-reduced pointers keep the hot loop
// to pointer bumps + loads + 4 WMMAs. Fixed-order LDS reduction in wave 0
// keeps the result bitwise deterministic; epilogue fuses bias + exact GELU.
__global__ __launch_bounds__(GEMM_NW * 32)
void gemm_wmma_f32_kernel(const float* __restrict__ A, int lda,
                          const float* __restrict__ B, int ldb,
                          const float* __restrict__ bias,
                          float* __restrict__ C, int ldc,
                          int N, int K, int do_gelu) {
    constexpr int NW = GEMM_NW;
    extern __shared__ float smem[];     // (NW-1) * 4 tiles * 8 regs * 32 lanes
    const int lane = threadIdx.x & 31;
    const int w    = threadIdx.x >> 5;  // wave id in block
    const int nn = lane & 15;           // N within tile / M row selector
    const int hi = lane >> 4;           // K-pair selector (0 or 1)
    const int n0 = blockIdx.x * 16;
    const int col = n0 + nn;
    const bool colOK = (col < N);
    const int colC = colOK ? col : (N - 1);   // clamped: garbage only feeds
                                              // this lane's (unstored) column
    v8f acc[4] = {v8f{}, v8f{}, v8f{}, v8f{}};

    const int cfull = K >> 2;                 // number of complete 4-deep chunks
    const int iters = (cfull > w) ? (cfull - w + NW - 1) / NW : 0;
    const int kb0   = 4 * w + 2 * hi;         // this wave's first K pair

    // strength-reduced streams: 2 B pointers (rows kb, kb+1), 4 A pointers
    const float* bp0 = B + (size_t)kb0 * ldb + colC;
    const float* bp1 = bp0 + ldb;
    const float* ap0 = A + (size_t)(nn     ) * lda + kb0;
    const float* ap1 = A + (size_t)(nn + 16) * lda + kb0;
    const float* ap2 = A + (size_t)(nn + 32) * lda + kb0;
    const float* ap3 = A + (size_t)(nn + 48) * lda + kb0;
    const size_t bstep = (size_t)(4 * NW) * ldb;
    const size_t astep = (size_t)(4 * NW);

    for (int i = 0; i < iters; ++i) {
        if (i + 8 < iters)                          // prefetch weight stream
            __builtin_prefetch(bp0 + 8 * bstep, 0, 1);
        v2f bf;  bf.x = bp0[0];  bf.y = bp1[0];
        v2f a0;  a0.x = ap0[0];  a0.y = ap0[1];
        v2f a1;  a1.x = ap1[0];  a1.y = ap1[1];
        v2f a2;  a2.x = ap2[0];  a2.y = ap2[1];
        v2f a3;  a3.x = ap3[0];  a3.y = ap3[1];
        acc[0] = __builtin_amdgcn_wmma_f32_16x16x4_f32(false, a0, false, bf,
                                                       (short)0, acc[0], false, false);
        acc[1] = __builtin_amdgcn_wmma_f32_16x16x4_f32(false, a1, false, bf,
                                                       (short)0, acc[1], false, false);
        acc[2] = __builtin_amdgcn_wmma_f32_16x16x4_f32(false, a2, false, bf,
                                                       (short)0, acc[2], false, false);
        acc[3] = __builtin_amdgcn_wmma_f32_16x16x4_f32(false, a3, false, bf,
                                                       (short)0, acc[3], false, false);
        bp0 += bstep; bp1 += bstep;
        ap0 += astep; ap1 += astep; ap2 += astep; ap3 += astep;
    }

    // K tail (K % 4 != 0): owned by the wave whose turn chunk `cfull` would be
    if ((K & 3) && (cfull % NW) == w) {
        const int kb = 4 * cfull + 2 * hi;
        v2f bf;
        bf.x = (kb     < K) ? B[(size_t)kb * ldb + colC] : 0.0f;
        bf.y = (kb + 1 < K) ? B[(size_t)(kb + 1) * ldb + colC] : 0.0f;
#pragma unroll
        for (int mt = 0; mt < 4; ++mt) {
            const int row = mt * 16 + nn;
            v2f af;
            af.x = (kb     < K) ? A[(size_t)row * lda + kb] : 0.0f;
            af.y = (kb + 1 < K) ? A[(size_t)row * lda + kb + 1] : 0.0f;
            acc[mt] = __builtin_amdgcn_wmma_f32_16x16x4_f32(
                false, af, false, bf, (short)0, acc[mt], false, false);
        }
    }

    // spill waves 1..NW-1, then fixed-order reduction in wave 0 (deterministic)
    if (w > 0) {
#pragma unroll
        for (int mt = 0; mt < 4; ++mt)
#pragma unroll
            for (int r = 0; r < 8; ++r)
                smem[(((size_t)(w - 1) * 4 + mt) * 8 + r) * 32 + lane] = acc[mt][r];
    }
    __syncthreads();
    if (w == 0) {
#pragma unroll 1
        for (int w2 = 0; w2 < NW - 1; ++w2)
#pragma unroll
            for (int mt = 0; mt < 4; ++mt)
#pragma unroll
                for (int r = 0; r < 8; ++r)
                    acc[mt][r] += smem[(((size_t)w2 * 4 + mt) * 8 + r) * 32 + lane];

        const float bv = colOK ? bias[col] : 0.0f;
#pragma unroll
        for (int mt = 0; mt < 4; ++mt) {
            const int mbase = mt * 16 + hi * 8;   // C/D layout: hi lanes hold M+8
#pragma unroll
            for (int r = 0; r < 8; ++r) {
                float v = acc[mt][r] + bv;
                if (do_gelu) v = gelu_exact(v);
                if (colOK) C[(size_t)(mbase + r) * ldc + col] = v;
            }
        }
    }
}

extern "C" void kernel_launch(void* const* d_in, const int* in_sizes, int n_in,
                              void* d_out, int out_size, void* d_ws, size_t ws_size,
                              hipStream_t stream) {
    (void)in_sizes; (void)n_in; (void)out_size; (void)ws_size;
    const int*   tags  = (const int*)  d_in[0];
    const float* fc    = (const float*)d_in[1];
    const float* ln_g  = (const float*)d_in[2];
    const float* ln_b  = (const float*)d_in[3];
    const float* w1    = (const float*)d_in[4];
    const float* b1    = (const float*)d_in[5];
    const float* ln2_g = (const float*)d_in[6];
    const float* ln2_b = (const float*)d_in[7];
    const float* we1   = (const float*)d_in[8];
    const float* be1   = (const float*)d_in[9];
    const float* we2   = (const float*)d_in[10];
    const float* be2   = (const float*)d_in[11];
    const float* wd1   = (const float*)d_in[12];
    const float* bd1   = (const float*)d_in[13];
    const float* wd2   = (const float*)d_in[14];
    const float* bd2   = (const float*)d_in[15];

    float* out = (float*)d_out;
    float* y   = out;                         // [64, 20001]
    float* enc = out + (size_t)BATCH * YW;    // [64, 256]
    float* dec = enc + (size_t)BATCH * HDIM;  // [64, 20001]
    float* cnt = dec;                         // scratch [64, 20000] (consumed
                                              // before dec is produced)

    float* h1p = (float*)d_ws;                // [64, 1024] pre-LN
    float* h1  = h1p + (size_t)BATCH * H4;    // [64, 1024]
    float* h2  = h1  + (size_t)BATCH * H4;    // [64, 1024]
    float* d3  = h2  + (size_t)BATCH * H4;    // [64, 1024]

    const int th = GEMM_NW * 32;
    const size_t sm = GEMM_SMEM;

    // 1. zero the tag-count scratch
    {
        int n = BATCH * NUM_TAGS;
        zero_f32_kernel<<<(n + 255) / 256, 256, 0, stream>>>(cnt, n);
    }
    // 2. dedup scatter-count
    scatter_count_kernel<<<(BATCH * NFEAT + 127) / 128, 128, 0, stream>>>(tags, cnt);
    // 3. LayerNorm(20000) + concat(feature_counts/100) -> y
    ln_y_kernel<<<BATCH, 256, 0, stream>>>(cnt, fc, ln_g, ln_b, y);
    // 4. h1p = gelu(y @ w1 + b1)          [64,20001]x[20001,1024]
    gemm_wmma_f32_kernel<<<H4 / 16, th, sm, stream>>>(y, YW, w1, H4, b1,
                                                      h1p, H4, H4, YW, 1);
    // 5. h1 = LayerNorm(h1p)
    ln_h_kernel<<<BATCH, 256, 0, stream>>>(h1p, ln2_g, ln2_b, h1);
    // 6. h2 = gelu(h1 @ we1 + be1)        [64,1024]x[1024,1024]
    gemm_wmma_f32_kernel<<<H4 / 16, th, sm, stream>>>(h1, H4, we1, H4, be1,
                                                      h2, H4, H4, H4, 1);
    // 7. enc = h2 @ we2 + be2             [64,1024]x[1024,256]
    gemm_wmma_f32_kernel<<<HDIM / 16, th, sm, stream>>>(h2, H4, we2, HDIM, be2,
                                                        enc, HDIM, HDIM, H4, 0);
    // 8. d3 = gelu(enc @ wd1 + bd1)       [64,256]x[256,1024]
    gemm_wmma_f32_kernel<<<H4 / 16, th, sm, stream>>>(enc, HDIM, wd1, H4, bd1,
                                                      d3, H4, H4, HDIM, 1);
    // 9. dec = d3 @ wd2 + bd2             [64,1024]x[1024,20001]
    gemm_wmma_f32_kernel<<<(YW + 15) / 16, th, sm, stream>>>(d3, H4, wd2, YW, bd2,
                                                             dec, YW, YW, H4, 0);
}